// DCTFFN_81535659147930
// MI455X (gfx1250) — compile-verified
//
#include <hip/hip_runtime.h>

typedef __attribute__((ext_vector_type(2))) float v2f;
typedef __attribute__((ext_vector_type(8))) float v8f;

constexpr int Hdim = 256, Wdim = 256, Cin = 64, C2 = 256, HID = 128, Cout = 64;
constexpr int HW = Hdim * Wdim;

// Orthonormal type-II DCT basis, A[k][n] = c_k * cos(pi*(2n+1)k/16)
__constant__ float DCT8[8][8] = {
  { 0.3535533906f, 0.3535533906f, 0.3535533906f, 0.3535533906f, 0.3535533906f, 0.3535533906f, 0.3535533906f, 0.3535533906f },
  { 0.4903926402f, 0.4157348062f, 0.2777851165f, 0.0975451610f,-0.0975451610f,-0.2777851165f,-0.4157348062f,-0.4903926402f },
  { 0.4619397663f, 0.1913417162f,-0.1913417162f,-0.4619397663f,-0.4619397663f,-0.1913417162f, 0.1913417162f, 0.4619397663f },
  { 0.4157348062f,-0.0975451610f,-0.4903926402f,-0.2777851165f, 0.2777851165f, 0.4903926402f, 0.0975451610f,-0.4157348062f },
  { 0.3535533906f,-0.3535533906f,-0.3535533906f, 0.3535533906f, 0.3535533906f,-0.3535533906f,-0.3535533906f, 0.3535533906f },
  { 0.2777851165f,-0.4903926402f, 0.0975451610f, 0.4157348062f,-0.4157348062f,-0.0975451610f, 0.4903926402f,-0.2777851165f },
  { 0.1913417162f,-0.4619397663f, 0.4619397663f,-0.1913417162f,-0.1913417162f, 0.4619397663f,-0.4619397663f, 0.1913417162f },
  { 0.0975451610f,-0.2777851165f, 0.4157348062f,-0.4903926402f, 0.4903926402f,-0.4157348062f, 0.2777851165f,-0.0975451610f }
};

// ---------------------------------------------------------------------------
// Kernel 1: project_in (WMMA fp32 GEMM, 256x64 weights) fused with the
// per-8x8-patch DCT -> mask -> second transform (as the reference einsums
// literally specify: both are A * X * A^T with the mask in between).
// Workgroup: 16 out-channels x 16x16 pixel tile (2x2 patches), 8 waves.
// ---------------------------------------------------------------------------
__global__ __launch_bounds__(256) void
k1_projin_dct(const float* __restrict__ x, const float* __restrict__ Win,
              const float* __restrict__ mix, float* __restrict__ ws)
{
    __shared__ float yt[16][272];   // [ch][py*17 + px]
    __shared__ float vt[16][272];

    const int tid  = threadIdx.x;
    const int lane = tid & 31;
    const int wv   = tid >> 5;
    const int m    = lane & 15;     // M index (out channel) and N index (pixel col)
    const int hi   = lane >> 4;

    const int b  = blockIdx.z >> 4;
    const int ob = blockIdx.z & 15;           // out-channel block (16 channels)
    const int r0 = blockIdx.y * 16;
    const int c0 = blockIdx.x * 16;

    // ---- GEMM: Y[16 ch][256 px] = Win_tile[16x64] * X[64 x 256] ----
    const float* xb = x + (size_t)b * Cin * HW;
    const float* wr = Win + (size_t)(ob * 16 + m) * Cin;
    const int t0 = 2 * wv, t1 = 2 * wv + 1;   // the two pixel rows this wave owns
    const float* xp0 = xb + (size_t)(r0 + t0) * Wdim + c0 + m;
    const float* xp1 = xb + (size_t)(r0 + t1) * Wdim + c0 + m;

    v8f acc0 = {}, acc1 = {};
#pragma unroll
    for (int kk = 0; kk < Cin; kk += 4) {
        const int k0 = kk + 2 * hi;
        v2f a;  a.x  = wr[k0];                    a.y  = wr[k0 + 1];
        v2f b0; b0.x = xp0[(size_t)k0 * HW];      b0.y = xp0[(size_t)(k0 + 1) * HW];
        v2f b1; b1.x = xp1[(size_t)k0 * HW];      b1.y = xp1[(size_t)(k0 + 1) * HW];
        acc0 = __builtin_amdgcn_wmma_f32_16x16x4_f32(false, a, false, b0, (short)0, acc0, false, false);
        acc1 = __builtin_amdgcn_wmma_f32_16x16x4_f32(false, a, false, b1, (short)0, acc1, false, false);
    }
#pragma unroll
    for (int r = 0; r < 8; ++r) {               // D: vgpr r -> M = r + 8*hi, N = m
        yt[r + 8 * hi][t0 * 17 + m] = acc0[r];
        yt[r + 8 * hi][t1 * 17 + m] = acc1[r];
    }
    __syncthreads();

    // ---- Phase 1: V = (A * Xp * A^T) .* mask  (512 row-tasks) ----
    const float* mx = mix + (size_t)(ob * 16) * 64;
#pragma unroll 1
    for (int task = tid; task < 512; task += 256) {
        const int i  = task & 7;
        const int u  = task >> 3;
        const int ch = u >> 2;
        const int pr = ((u >> 1) & 1) * 8;
        const int pc = (u & 1) * 8;
        float U[8];
#pragma unroll
        for (int j = 0; j < 8; ++j) {
            float s = 0.f;
#pragma unroll
            for (int k = 0; k < 8; ++k) s += DCT8[i][k] * yt[ch][(pr + k) * 17 + pc + j];
            U[j] = s;
        }
#pragma unroll
        for (int q = 0; q < 8; ++q) {
            float s = 0.f;
#pragma unroll
            for (int j = 0; j < 8; ++j) s += U[j] * DCT8[q][j];
            vt[ch][(pr + i) * 17 + pc + q] = s * mx[ch * 64 + i * 8 + q];
        }
    }
    __syncthreads();

    // ---- Phase 2: out = A * V * A^T (reference's 'ip,..pq,jq' einsum) ----
#pragma unroll 1
    for (int task = tid; task < 512; task += 256) {
        const int i  = task & 7;
        const int u  = task >> 3;
        const int ch = u >> 2;
        const int pr = ((u >> 1) & 1) * 8;
        const int pc = (u & 1) * 8;
        float U[8];
#pragma unroll
        for (int j = 0; j < 8; ++j) {
            float s = 0.f;
#pragma unroll
            for (int k = 0; k < 8; ++k) s += DCT8[i][k] * vt[ch][(pr + k) * 17 + pc + j];
            U[j] = s;
        }
#pragma unroll
        for (int q = 0; q < 8; ++q) {
            float s = 0.f;
#pragma unroll
            for (int j = 0; j < 8; ++j) s += U[j] * DCT8[q][j];
            yt[ch][(pr + i) * 17 + pc + q] = s;
        }
    }
    __syncthreads();

    // ---- Coalesced store to workspace [B][256][H][W] ----
    const int py = tid >> 4, px = tid & 15;
    size_t obase = ((size_t)b * C2 + ob * 16) * HW + (size_t)(r0 + py) * Wdim + (c0 + px);
#pragma unroll
    for (int ch = 0; ch < 16; ++ch)
        ws[obase + (size_t)ch * HW] = yt[ch][py * 17 + px];
}

// ---------------------------------------------------------------------------
// Kernel 2: depthwise 3x3 (SAME, cross-correlation) + tanh-GELU gate fused
// with project_out (WMMA fp32 GEMM, 64x128). Workgroup: 16x16 pixel tile,
// full 64-out-channel accumulator held in registers (8 waves x 8 v8f tiles),
// hidden dim processed in chunks of 16 staged through LDS.
// ---------------------------------------------------------------------------
__global__ __launch_bounds__(256) void
k2_dw_gate_out(const float* __restrict__ ws, const float* __restrict__ wdw,
               const float* __restrict__ wout, float* __restrict__ out)
{
    __shared__ float g[16][272];    // [local hidden ch][py*16 + px], stride 272 -> bank-split halves

    const int tid  = threadIdx.x;
    const int lane = tid & 31;
    const int wv   = tid >> 5;
    const int m    = lane & 15;
    const int hi   = lane >> 4;

    const int b  = blockIdx.z;
    const int r0 = blockIdx.y * 16;
    const int c0 = blockIdx.x * 16;
    const int py = tid >> 4, px = tid & 15;
    const int gy = r0 + py,  gx = c0 + px;

    const float* wsb  = ws + (size_t)b * C2 * HW;
    const int mt  = wv >> 1;            // out-channel tile (4 tiles of 16)
    const int ntb = (wv & 1) * 8;       // pixel-row tiles: 8 per wave
    const float* wrow = wout + (size_t)(mt * 16 + m) * HID;

    v8f acc[8];
#pragma unroll
    for (int j = 0; j < 8; ++j) acc[j] = (v8f){};

    for (int kc = 0; kc < HID; kc += 16) {
        __syncthreads();                // previous chunk's B reads complete
        // depthwise conv + gate for hidden channels kc..kc+15
#pragma unroll 1
        for (int i = 0; i < 16; ++i) {
            const int c = kc + i;
            const float* w1 = wdw + c * 9;
            const float* w2 = wdw + (c + HID) * 9;
            const float* p1 = wsb + (size_t)c * HW;
            const float* p2 = wsb + (size_t)(c + HID) * HW;
            float s1 = 0.f, s2 = 0.f;
#pragma unroll
            for (int ky = 0; ky < 3; ++ky) {
                const int yy = gy + ky - 1;
                const bool iny = ((unsigned)yy < (unsigned)Hdim);
#pragma unroll
                for (int kx = 0; kx < 3; ++kx) {
                    const int xx = gx + kx - 1;
                    const bool in = iny && ((unsigned)xx < (unsigned)Wdim);
                    const size_t off = (size_t)yy * Wdim + xx;
                    const float v1 = in ? p1[off] : 0.f;
                    const float v2 = in ? p2[off] : 0.f;
                    s1 += w1[ky * 3 + kx] * v1;
                    s2 += w2[ky * 3 + kx] * v2;
                }
            }
            // jax.nn.gelu (approximate=True, tanh form)
            const float t = tanhf(0.7978845608028654f * (s1 + 0.044715f * s1 * s1 * s1));
            g[i][py * 16 + px] = 0.5f * s1 * (1.f + t) * s2;
        }
        __syncthreads();
        // WMMA: acc[mt, nt] += Wout[16 x 16k] * g[16k x 16px]
#pragma unroll
        for (int kk = 0; kk < 16; kk += 4) {
            const int k0 = kk + 2 * hi;
            v2f a; a.x = wrow[kc + k0]; a.y = wrow[kc + k0 + 1];
#pragma unroll
            for (int j = 0; j < 8; ++j) {
                const int nt = ntb + j;
                v2f bv; bv.x = g[k0][nt * 16 + m]; bv.y = g[k0 + 1][nt * 16 + m];
                acc[j] = __builtin_amdgcn_wmma_f32_16x16x4_f32(false, a, false, bv, (short)0, acc[j], false, false);
            }
        }
    }

    // ---- Store: out[B][64][H][W] ----
#pragma unroll
    for (int j = 0; j < 8; ++j) {
        const int nt = ntb + j;
        size_t base = ((size_t)b * Cout + mt * 16 + 8 * hi) * HW
                    + (size_t)(r0 + nt) * Wdim + c0 + m;
#pragma unroll
        for (int r = 0; r < 8; ++r)
            out[base + (size_t)r * HW] = acc[j][r];
    }
}

extern "C" void kernel_launch(void* const* d_in, const int* in_sizes, int n_in,
                              void* d_out, int out_size, void* d_ws, size_t ws_size,
                              hipStream_t stream) {
    const float* x    = (const float*)d_in[0];   // [2,64,256,256]
    const float* Win  = (const float*)d_in[1];   // [256,64]
    const float* Wdw  = (const float*)d_in[2];   // [256,1,3,3]
    const float* mix  = (const float*)d_in[3];   // [1,256,1,1,8,8]
    const float* Wout = (const float*)d_in[4];   // [64,128]
    float* out = (float*)d_out;                  // [2,64,256,256]
    float* ws  = (float*)d_ws;                   // needs 2*256*256*256*4 = 128 MB

    dim3 blk(256);
    dim3 g1(Wdim / 16, Hdim / 16, 2 * 16);
    k1_projin_dct<<<g1, blk, 0, stream>>>(x, Win, mix, ws);
    dim3 g2(Wdim / 16, Hdim / 16, 2);
    k2_dw_gate_out<<<g2, blk, 0, stream>>>(ws, Wdw, Wout, out);
}